// GDE_48515950575729
// MI455X (gfx1250) — compile-verified
//
#include <hip/hip_runtime.h>
#include <math.h>

// ---------------------------------------------------------------------------
// Graph Neural ODE (SplineConv encoder -> RK4(2x SplineConv) -> SplineConv
// decoder -> log_softmax) for MI455X (gfx1250, wave32).
//
// Dense transforms: V_WMMA_F32_16X16X4_F32 (fp32 matrix core), A tile staged
// via GLOBAL_LOAD_ASYNC_TO_LDS_B128 (ASYNCcnt). GEMM is templated on
// (Fin, Fout) so all B/W addresses fold into 24-bit VMEM immediate offsets.
// Edge gather/scatter: one wave32 per edge + global_atomic_add_f32 (L2).
// ---------------------------------------------------------------------------

typedef __attribute__((ext_vector_type(2))) float v2f;
typedef __attribute__((ext_vector_type(8))) float v8f;

#define NNODES 50000   // multiple of 16 -> no row-tile tail in the GEMM
#define FIN    128
#define HID    64
#define NCLS   47

#if __has_builtin(__builtin_amdgcn_global_load_async_to_lds_b128)
#define USE_ASYNC_LDS 1
// Exact parameter types per hipcc diagnostic: pointee is a 16-byte int vector,
// global side in addrspace(1), LDS side in addrspace(3).
typedef int v4i_b128 __attribute__((vector_size(16)));
typedef __attribute__((address_space(1))) v4i_b128 as1_v4i;
typedef __attribute__((address_space(3))) v4i_b128 as3_v4i;
#else
#define USE_ASYNC_LDS 0
#endif

// ---------------------------------------------------------------------------
// GEMM3: out0 = X@W0, out1 = X@W1, base = X@Wr + bias   (shared A operand)
// Block = 128 threads (4 waves), one 16-row tile of X staged in LDS.
// Wave w owns column tiles w, w+4, ...; 3 WMMA accumulators per tile.
// A-frag (f32 16x4, ISA 7.12.2): lanes 0-15 -> (K0,K1), lanes 16-31 -> (K2,K3).
// B-frag (f32 4x16, symmetric): v0 = rows {0,2}, v1 = rows {1,3}, col = lane&15.
// C/D: VGPR v of lane l -> row v + 8*(l>>4), col = l&15.
// Out-of-range columns (Fout=47) are CLAMPED for loads (lane n's B column only
// feeds output column n) and masked only at store time -> convergent K loop.
// TFIN/TFOUT compile-time -> all W/A/store offsets become immediates.
// ---------------------------------------------------------------------------
template <int TFIN, int TFOUT>
__global__ void gemm3_wmma_kernel(const float* __restrict__ X,
                                  const float* __restrict__ W0,
                                  const float* __restrict__ W1,
                                  const float* __restrict__ Wr,
                                  const float* __restrict__ bias,
                                  float* __restrict__ out0,
                                  float* __restrict__ out1,
                                  float* __restrict__ base)
{
    constexpr int STRIDE_A = TFIN + 4;               // +4 pad: conflict-free LDS
    __shared__ float ldsA[16 * STRIDE_A];
    const int row0 = blockIdx.x * 16;
    const int tid  = threadIdx.x;                    // 0..127

    // ---- stage the 16 x TFIN A tile into LDS (16B chunks, coalesced) ----
    constexpr int nChunks = (16 * TFIN) / 4;         // float4 chunks
    constexpr int cPerRow = TFIN / 4;
#if USE_ASYNC_LDS
    #pragma unroll
    for (int ci = tid; ci < nChunks; ci += 128) {
        int r = ci / cPerRow, c4 = (ci % cPerRow) << 2;
        const float* g = X + (size_t)(row0 + r) * TFIN + c4;
        float*       l = &ldsA[r * STRIDE_A + c4];
        __builtin_amdgcn_global_load_async_to_lds_b128((as1_v4i*)g, (as3_v4i*)l, 0, 0);
    }
#if __has_builtin(__builtin_amdgcn_s_wait_asynccnt)
    __builtin_amdgcn_s_wait_asynccnt(0);
#else
    asm volatile("s_wait_asynccnt 0x0" ::: "memory");
#endif
#else
    #pragma unroll
    for (int ci = tid; ci < nChunks; ci += 128) {
        int r = ci / cPerRow, c4 = (ci % cPerRow) << 2;
        *(float4*)(&ldsA[r * STRIDE_A + c4]) =
            *(const float4*)(X + (size_t)(row0 + r) * TFIN + c4);
    }
#endif
    __syncthreads();

    const int lane = tid & 31;
    const int wave = tid >> 5;                       // 0..3
    const int half = lane >> 4;                      // 0 | 1
    const int l16  = lane & 15;
    constexpr int nColTiles = (TFOUT + 15) / 16;

    for (int ct = wave; ct < nColTiles; ct += 4) {
        const int  col  = ct * 16 + l16;
        const bool cok  = (col < TFOUT);
        const int  colc = cok ? col : (TFOUT - 1);   // clamp: convergent K loop

        // Per-lane weight base pointers; all K offsets are compile-time
        // immediates (k*TFOUT*4 <= 32KB < 24-bit IOFFSET).
        const float* pA  = &ldsA[l16 * STRIDE_A + 2 * half];
        const float* p0  = W0 + (size_t)(2 * half) * TFOUT + colc;
        const float* p1  = W1 + (size_t)(2 * half) * TFOUT + colc;
        const float* pr  = Wr + (size_t)(2 * half) * TFOUT + colc;

        v8f acc0 = {}; v8f acc1 = {}; v8f accr = {};
        #pragma unroll 4
        for (int k0 = 0; k0 < TFIN; k0 += 4) {
            v2f a;
            a.x = pA[k0];
            a.y = pA[k0 + 1];

            v2f b0, b1, br;
            b0.x = p0[k0 * TFOUT];  b0.y = p0[(k0 + 1) * TFOUT];
            b1.x = p1[k0 * TFOUT];  b1.y = p1[(k0 + 1) * TFOUT];
            br.x = pr[k0 * TFOUT];  br.y = pr[(k0 + 1) * TFOUT];

            acc0 = __builtin_amdgcn_wmma_f32_16x16x4_f32(false, a, false, b0,
                                                         (short)0, acc0, false, false);
            acc1 = __builtin_amdgcn_wmma_f32_16x16x4_f32(false, a, false, b1,
                                                         (short)0, acc1, false, false);
            accr = __builtin_amdgcn_wmma_f32_16x16x4_f32(false, a, false, br,
                                                         (short)0, accr, false, false);
        }

        if (cok) {
            const float bb = bias[col];
            float* q0 = out0 + (size_t)(row0 + 8 * half) * TFOUT + col;
            float* q1 = out1 + (size_t)(row0 + 8 * half) * TFOUT + col;
            float* qb = base + (size_t)(row0 + 8 * half) * TFOUT + col;
            #pragma unroll
            for (int v = 0; v < 8; ++v) {
                q0[v * TFOUT] = acc0[v];
                q1[v * TFOUT] = acc1[v];
                qb[v * TFOUT] = accr[v] + bb;
            }
        }
    }
}

// ---------------------------------------------------------------------------
// Edge degree (once; depends only on dst), then inverted in place.
// ---------------------------------------------------------------------------
__global__ void degree_kernel(const int* __restrict__ dst, float* __restrict__ deg,
                              int nEdges)
{
    int e = blockIdx.x * blockDim.x + threadIdx.x;
    if (e < nEdges) unsafeAtomicAdd(&deg[dst[e]], 1.0f);
}

__global__ void invdeg_kernel(float* __restrict__ deg, int n)
{
    int i = blockIdx.x * blockDim.x + threadIdx.x;
    if (i < n) deg[i] = 1.0f / fmaxf(deg[i], 1.0f);
}

// ---------------------------------------------------------------------------
// Edge message + scatter: one wave32 per edge.
//   msg = (1-u)*X0[src] + u*X1[src];  agg[dst] += msg  (global_atomic_add_f32)
// H==64: each lane handles one float2 (b64 coalesced gathers, 2 atomics).
// ---------------------------------------------------------------------------
__global__ void edge_scatter_kernel(const int* __restrict__ src,
                                    const int* __restrict__ dst,
                                    const float* __restrict__ u,
                                    const float* __restrict__ X0,
                                    const float* __restrict__ X1,
                                    float* __restrict__ agg,
                                    int H, int nEdges)
{
    int wid  = (blockIdx.x * blockDim.x + threadIdx.x) >> 5;
    int lane = threadIdx.x & 31;
    if (wid >= nEdges) return;
    int   s  = src[wid];
    int   d  = dst[wid];
    float ue = u[wid];
    float w0 = 1.f - ue;
    float* ad = agg + (size_t)d * H;

    if (H == 64) {
        const float2 a = ((const float2*)(X0 + ((size_t)s << 6)))[lane];
        const float2 b = ((const float2*)(X1 + ((size_t)s << 6)))[lane];
        unsafeAtomicAdd(&ad[2 * lane],     w0 * a.x + ue * b.x);
        unsafeAtomicAdd(&ad[2 * lane + 1], w0 * a.y + ue * b.y);
    } else {
        const float* xs0 = X0 + (size_t)s * H;
        const float* xs1 = X1 + (size_t)s * H;
        for (int j = lane; j < H; j += 32)
            unsafeAtomicAdd(&ad[j], w0 * xs0[j] + ue * xs1[j]);
    }
}

// ---------------------------------------------------------------------------
// out[i] = act( agg[i]*invdeg[node] + out[i] )    (base pre-stored in out)
// ---------------------------------------------------------------------------
__global__ void combine_kernel(const float* __restrict__ agg,
                               const float* __restrict__ invdeg,
                               float* __restrict__ out,
                               int H, int n, int do_tanh)
{
    int i = blockIdx.x * blockDim.x + threadIdx.x;
    if (i >= n * H) return;
    int   node = i / H;
    float v    = fmaf(agg[i], invdeg[node], out[i]);
    out[i] = do_tanh ? tanhf(v) : v;
}

// acc += w*k ;  optionally y = h + c*k
__global__ void rk_update_kernel(const float* __restrict__ k,
                                 const float* __restrict__ h,
                                 float* __restrict__ acc,
                                 float* __restrict__ y,
                                 float w, float c, int n, int write_y)
{
    int i = blockIdx.x * blockDim.x + threadIdx.x;
    if (i >= n) return;
    float kv = k[i];
    acc[i] += w * kv;
    if (write_y) y[i] = fmaf(c, kv, h[i]);
}

// h += s * acc
__global__ void axpy_kernel(float* __restrict__ h, const float* __restrict__ acc,
                            float s, int n)
{
    int i = blockIdx.x * blockDim.x + threadIdx.x;
    if (i < n) h[i] = fmaf(s, acc[i], h[i]);
}

// ---------------------------------------------------------------------------
// Row log-softmax over C classes: one wave32 per row, shuffle reductions.
// ---------------------------------------------------------------------------
__global__ void logsoftmax_kernel(const float* __restrict__ in,
                                  float* __restrict__ out, int C, int n)
{
    int node = (blockIdx.x * blockDim.x + threadIdx.x) >> 5;
    int lane = threadIdx.x & 31;
    if (node >= n) return;
    const float* r = in + (size_t)node * C;
    float m = -INFINITY;
    for (int j = lane; j < C; j += 32) m = fmaxf(m, r[j]);
    #pragma unroll
    for (int o = 16; o; o >>= 1) m = fmaxf(m, __shfl_xor(m, o, 32));
    float s = 0.f;
    for (int j = lane; j < C; j += 32) s += __expf(r[j] - m);
    #pragma unroll
    for (int o = 16; o; o >>= 1) s += __shfl_xor(s, o, 32);
    float lse = __logf(s) + m;
    float* w = out + (size_t)node * C;
    for (int j = lane; j < C; j += 32) w[j] = r[j] - lse;
}

// ---------------------------------------------------------------------------
// Host orchestration
// ---------------------------------------------------------------------------
extern "C" void kernel_launch(void* const* d_in, const int* in_sizes, int n_in,
                              void* d_out, int out_size, void* d_ws, size_t ws_size,
                              hipStream_t stream)
{
    const float* x      = (const float*)d_in[0];
    const int*   eidx   = (const int*)d_in[1];        // [2, E] (src row, dst row)
    const float* u      = (const float*)d_in[2];      // [E]
    const float* W_in   = (const float*)d_in[3];      // [2,128,64]
    const float* Wr_in  = (const float*)d_in[4];
    const float* b_in   = (const float*)d_in[5];
    const float* W_fa   = (const float*)d_in[6];      // [2,64,64]
    const float* Wr_fa  = (const float*)d_in[7];
    const float* b_fa   = (const float*)d_in[8];
    const float* W_fb   = (const float*)d_in[9];
    const float* Wr_fb  = (const float*)d_in[10];
    const float* b_fb   = (const float*)d_in[11];
    const float* W_out  = (const float*)d_in[12];     // [2,64,47]
    const float* Wr_out = (const float*)d_in[13];
    const float* b_out  = (const float*)d_in[14];
    float*       out    = (float*)d_out;

    const int N = NNODES;
    const int E = in_sizes[2];
    const int* src = eidx;
    const int* dst = eidx + E;

    // ---- workspace layout (floats) ----
    const size_t NH   = (size_t)N * HID;              // 3.2M floats
    const size_t NPAD = 50048;                        // deg slot, 16B aligned
    float* ws  = (float*)d_ws;
    float* deg = ws;                                  // holds 1/max(deg,1)
    float* X0  = ws + NPAD;
    float* X1  = X0 + NH;
    float* agg = X1 + NH;
    float* h   = agg + NH;
    float* h1  = h + NH;                              // mid / decoder buffer
    float* y   = h1 + NH;
    float* k   = y + NH;
    float* acc = k + NH;

    const int rowTiles  = N / 16;                     // N % 16 == 0
    const int edgeBlk   = (E * 32 + 255) / 256;       // one wave per edge
    const int elemBlkH  = ((int)NH + 255) / 256;
    const size_t aggBytes = NH * sizeof(float);

    // ---- inverse degree (once) ----
    (void)hipMemsetAsync(deg, 0, N * sizeof(float), stream);
    degree_kernel<<<(E + 255) / 256, 256, 0, stream>>>(dst, deg, E);
    invdeg_kernel<<<(N + 255) / 256, 256, 0, stream>>>(deg, N);

    // ---- conv helper (compile-time Fin/Fout) ----
    auto post = [&](float* dest, int Fout, int do_tanh) {
        (void)hipMemsetAsync(agg, 0, aggBytes, stream);
        edge_scatter_kernel<<<edgeBlk, 256, 0, stream>>>(src, dst, u, X0, X1,
                                                         agg, Fout, E);
        int nb = (N * Fout + 255) / 256;
        combine_kernel<<<nb, 256, 0, stream>>>(agg, deg, dest, Fout, N, do_tanh);
    };
    auto conv_in = [&](const float* Xin, float* dest) {
        gemm3_wmma_kernel<FIN, HID><<<rowTiles, 128, 0, stream>>>(
            Xin, W_in, W_in + FIN * HID, Wr_in, b_in, X0, X1, dest);
        post(dest, HID, 1);
    };
    auto conv_hid = [&](const float* Xin, const float* W, const float* Wr,
                        const float* b, float* dest) {
        gemm3_wmma_kernel<HID, HID><<<rowTiles, 128, 0, stream>>>(
            Xin, W, W + HID * HID, Wr, b, X0, X1, dest);
        post(dest, HID, 0);
    };
    auto conv_out = [&](const float* Xin, float* dest) {
        gemm3_wmma_kernel<HID, NCLS><<<rowTiles, 128, 0, stream>>>(
            Xin, W_out, W_out + HID * NCLS, Wr_out, b_out, X0, X1, dest);
        post(dest, NCLS, 1);
    };

    // ---- encoder: h = tanh(conv_in(x)) ----
    conv_in(x, h);

    // ---- ODE vector field: func(in)->kout = conv_fb(conv_fa(in)) ----
    auto func = [&](const float* in, float* kout) {
        conv_hid(in, W_fa, Wr_fa, b_fa, h1);
        conv_hid(h1, W_fb, Wr_fb, b_fb, kout);
    };

    // ---- single RK4 step over [0, 1] ----
    (void)hipMemsetAsync(acc, 0, aggBytes, stream);
    func(h, k);   // k1
    rk_update_kernel<<<elemBlkH, 256, 0, stream>>>(k, h, acc, y, 1.f, 0.5f, (int)NH, 1);
    func(y, k);   // k2
    rk_update_kernel<<<elemBlkH, 256, 0, stream>>>(k, h, acc, y, 2.f, 0.5f, (int)NH, 1);
    func(y, k);   // k3
    rk_update_kernel<<<elemBlkH, 256, 0, stream>>>(k, h, acc, y, 2.f, 1.0f, (int)NH, 1);
    func(y, k);   // k4
    rk_update_kernel<<<elemBlkH, 256, 0, stream>>>(k, h, acc, y, 1.f, 0.0f, (int)NH, 0);
    axpy_kernel<<<elemBlkH, 256, 0, stream>>>(h, acc, 1.0f / 6.0f, (int)NH);

    // ---- decoder: dec = tanh(conv_out(h)); out = log_softmax(dec) ----
    conv_out(h, h1);
    logsoftmax_kernel<<<(N * 32 + 255) / 256, 256, 0, stream>>>(h1, out, NCLS, N);
}